// GIFNeuron_19215683682876
// MI455X (gfx1250) — compile-verified
//
#include <hip/hip_runtime.h>

typedef __attribute__((ext_vector_type(2))) float v2f;
typedef __attribute__((ext_vector_type(8))) float v8f;

// Problem dimensions (fixed by the reference)
#define BB 8
#define SS 4096
#define DD 512
#define HH 512
#define MM (BB * SS)          // 32768 rows of the GEMM

#define DECAY_F   0.90483741803595957f   // exp(-1/10)
#define ALPHA_F   0.01f
#define ONE_MA_F  0.99f                  // 1 - ALPHA
#define EPS_F     1e-6f
#define LMAX_F    16.0f                  // L
#define CLAMP_K   32.0f                  // L * 2

// ---------------------------------------------------------------------------
// Kernel 1: h = x @ W^T + b  via V_WMMA_F32_16X16X4_F32.
// Each wave computes a 16(M) x 64(N) strip: 4 accumulators, A reused 4x.
// Grid: 2048 tiles in M x 8 strips in N = 16384 waves = 2048 blocks of 8 waves.
// ---------------------------------------------------------------------------
__global__ __launch_bounds__(256) void gif_gemm_wmma(
    const float* __restrict__ X,   // (M, D) = (32768, 512)
    const float* __restrict__ Wm,  // (H, D) = (512, 512)
    const float* __restrict__ bias,// (H)
    float* __restrict__ Hout)      // (M, H) written in-place into d_out
{
    const int lane = threadIdx.x & 31;
    const int wave = threadIdx.x >> 5;
    const int gw   = blockIdx.x * 8 + wave;       // global wave id
    const int tileM = gw >> 3;                    // 2048 M-tiles
    const int tileN = (gw & 7) * 64;              // 8 N-strips of 64
    const int l16  = lane & 15;
    const int hi   = lane >> 4;                   // lanes 16-31 carry K+2 / M+8

    // A fragment: row = tileM*16 + l16, two consecutive K values per lane.
    const float* arow = X + (tileM * 16 + l16) * DD + hi * 2;
    // B fragments: col n within strip, element W[col, k+kk] (B = W^T slice).
    const float* brow0 = Wm + (tileN +  0 + l16) * DD + hi * 2;
    const float* brow1 = Wm + (tileN + 16 + l16) * DD + hi * 2;
    const float* brow2 = Wm + (tileN + 32 + l16) * DD + hi * 2;
    const float* brow3 = Wm + (tileN + 48 + l16) * DD + hi * 2;

    v8f acc0 = {}; v8f acc1 = {}; v8f acc2 = {}; v8f acc3 = {};

#pragma unroll 4
    for (int k = 0; k < DD; k += 4) {
        v2f a  = *(const v2f*)(arow  + k);
        v2f w0 = *(const v2f*)(brow0 + k);
        v2f w1 = *(const v2f*)(brow1 + k);
        v2f w2 = *(const v2f*)(brow2 + k);
        v2f w3 = *(const v2f*)(brow3 + k);
        acc0 = __builtin_amdgcn_wmma_f32_16x16x4_f32(false, a, false, w0,
                                                     (short)0, acc0, false, false);
        acc1 = __builtin_amdgcn_wmma_f32_16x16x4_f32(false, a, false, w1,
                                                     (short)0, acc1, false, false);
        acc2 = __builtin_amdgcn_wmma_f32_16x16x4_f32(false, a, false, w2,
                                                     (short)0, acc2, false, false);
        acc3 = __builtin_amdgcn_wmma_f32_16x16x4_f32(false, a, false, w3,
                                                     (short)0, acc3, false, false);
    }

    // C/D layout: lane l16 holds column N=l16; VGPR r holds row M = r + hi*8.
    float* outp = Hout + (tileM * 16 + hi * 8) * HH + tileN + l16;
    const float b0 = bias[tileN +  0 + l16];
    const float b1 = bias[tileN + 16 + l16];
    const float b2 = bias[tileN + 32 + l16];
    const float b3 = bias[tileN + 48 + l16];
#pragma unroll
    for (int r = 0; r < 8; ++r) {
        outp[r * HH +  0] = acc0[r] + b0;
        outp[r * HH + 16] = acc1[r] + b1;
        outp[r * HH + 32] = acc2[r] + b2;
        outp[r * HH + 48] = acc3[r] + b3;
    }
}

// ---------------------------------------------------------------------------
// Kernel 2: serial GIF scan. One thread per (b, h) chain; h values are read
// from `buf` and overwritten in place by the spike outputs. Loads are
// software-pipelined 8 timesteps ahead of the recurrence.
// ---------------------------------------------------------------------------
__global__ __launch_bounds__(256) void gif_scan(
    float* __restrict__ buf,    // (B, S, H): h in, spikes out (in place)
    float* __restrict__ vout,   // (B, H)
    float* __restrict__ thout)  // (B, H)
{
    const int tid = blockIdx.x * blockDim.x + threadIdx.x;   // 0..4095
    const int b   = tid >> 9;          // /512
    const int h   = tid & (HH - 1);
    float* p = buf + (size_t)b * SS * HH + h;

    float v = 0.0f;
    float theta = 1.0f;

    float cur[8];
#pragma unroll
    for (int i = 0; i < 8; ++i) cur[i] = p[i * HH];

    for (int t0 = 0; t0 < SS; t0 += 8) {
        float nxt[8];
        if (t0 + 8 < SS) {
#pragma unroll
            for (int i = 0; i < 8; ++i) nxt[i] = p[(t0 + 8 + i) * HH];
        } else {
#pragma unroll
            for (int i = 0; i < 8; ++i) nxt[i] = 0.0f;
        }
#pragma unroll
        for (int i = 0; i < 8; ++i) {
            v = fmaf(v, DECAY_F, cur[i]);
            const float lim = CLAMP_K * theta;            // L * theta * 2
            v = fminf(fmaxf(v, -lim), lim);               // -> v_med3 clamp
            const float norm = v * __builtin_amdgcn_rcpf(theta + EPS_F);
            float sp = floorf(norm);
            sp = fminf(fmaxf(sp, 0.0f), LMAX_F);          // spike in [0, L]
            v = fmaf(-sp, theta, v);                      // uses old theta
            theta = fmaf(ONE_MA_F, theta, fmaf(ALPHA_F, sp, ALPHA_F));
            p[(t0 + i) * HH] = sp;                        // overwrite h -> spike
        }
#pragma unroll
        for (int i = 0; i < 8; ++i) cur[i] = nxt[i];
    }

    vout[tid]  = v;
    thout[tid] = theta;
}

// ---------------------------------------------------------------------------
extern "C" void kernel_launch(void* const* d_in, const int* in_sizes, int n_in,
                              void* d_out, int out_size, void* d_ws, size_t ws_size,
                              hipStream_t stream) {
    (void)in_sizes; (void)n_in; (void)out_size; (void)d_ws; (void)ws_size;

    const float* x    = (const float*)d_in[0];   // (8, 4096, 512)
    const float* W    = (const float*)d_in[1];   // (512, 512)
    const float* bias = (const float*)d_in[2];   // (512)

    float* out   = (float*)d_out;
    float* hbuf  = out;                               // spikes region doubles as h
    float* vout  = out + (size_t)BB * SS * HH;        // +16777216
    float* thout = vout + (size_t)BB * HH;            // +4096

    // GEMM: 2048 M-tiles x 8 N-strips = 16384 waves = 2048 blocks of 256.
    gif_gemm_wmma<<<dim3(2048), dim3(256), 0, stream>>>(x, W, bias, hbuf);

    // Scan: 4096 chains, one per thread.
    gif_scan<<<dim3(16), dim3(256), 0, stream>>>(hbuf, vout, thout);
}